// dis_loss_27084063768796
// MI455X (gfx1250) — compile-verified
//
#include <hip/hip_runtime.h>
#include <cstdint>
#include <cstddef>

// ---------------------------------------------------------------------------
// mean over 262144 x 21 of ||pred-target||_2 (3 dims each).
// 132 MB read / 23.3 TB/s  =>  ~5.7 us floor: pure HBM-bandwidth problem.
// FLOPs (~5.5M sqrt, ~33M FMA) are noise -> matrix ops inapplicable.
// CDNA5 levers used: async global->LDS streaming (ASYNCcnt) with double
// buffering, 128-bit transactions, wave32 shfl reduction, deterministic
// two-kernel tree reduction (no atomics -> bitwise-stable output).
// ---------------------------------------------------------------------------

namespace {

constexpr int       kB              = 262144;
constexpr int       kJoints        = 21;
constexpr long long kTriples       = (long long)kB * kJoints;   // 5,505,024
constexpr int       kNT            = 256;   // 8 wave32 waves per block
constexpr int       kNB            = 672;   // 672 blocks * 8 chunks = 5376 chunks
constexpr int       kCPB           = 8;     // chunks per block
constexpr int       kQuadsPerChunk = 256;   // one quad (4 triples, 48 B) per thread
constexpr int       kFloatsPerThr  = 12;
constexpr int       kChunkFloats   = kQuadsPerChunk * kFloatsPerThr; // 3072
static_assert((long long)kNB * kCPB * kChunkFloats == kTriples * 3,
              "grid must tile the input exactly (no tail)");

typedef __attribute__((ext_vector_type(4))) float f32x4;
typedef __attribute__((ext_vector_type(4))) int   v4i;
typedef __attribute__((address_space(1))) v4i     gv4i;   // global int4
typedef __attribute__((address_space(3))) v4i     lv4i;   // LDS int4

#if defined(__HIP_DEVICE_COMPILE__)
#  if __has_builtin(__builtin_amdgcn_global_load_async_to_lds_b128)
#    define ASYNC_MODE 1
#  elif __has_builtin(__builtin_amdgcn_load_to_lds)
#    define ASYNC_MODE 2
#  elif __has_builtin(__builtin_amdgcn_global_load_lds)
#    define ASYNC_MODE 3
#  else
#    define ASYNC_MODE 0
#  endif
#else
#  define ASYNC_MODE 0
#endif

#if ASYNC_MODE
__device__ __forceinline__ void cp16(const float* g, float* l) {
  // chained casts: drop const -> generic void* -> generic v4i* -> AS-qualified
  gv4i* gp = (gv4i*)(v4i*)(void*)const_cast<float*>(g);
  lv4i* lp = (lv4i*)(v4i*)(void*)l;
#if ASYNC_MODE == 1
  // gfx1250: GLOBAL_LOAD_ASYNC_TO_LDS_B128 (tracked with ASYNCcnt)
  __builtin_amdgcn_global_load_async_to_lds_b128(gp, lp, /*offset=*/0, /*cpol=*/0);
#elif ASYNC_MODE == 2
  __builtin_amdgcn_load_to_lds(gp, lp, /*size=*/16, /*offset=*/0, /*cpol=*/0);
#else
  __builtin_amdgcn_global_load_lds(gp, lp, /*size=*/16, /*offset=*/0, /*cpol=*/0);
#endif
}

__device__ __forceinline__ void waitAsync6() {
#if __has_builtin(__builtin_amdgcn_s_wait_asynccnt)
  __builtin_amdgcn_s_wait_asynccnt(6);
#else
  asm volatile("s_wait_asynccnt 6" ::: "memory");
#endif
  asm volatile("" ::: "memory");   // keep LDS reads below the wait
}

__device__ __forceinline__ void waitAsync0() {
#if __has_builtin(__builtin_amdgcn_s_wait_asynccnt)
  __builtin_amdgcn_s_wait_asynccnt(0);
#else
  asm volatile("s_wait_asynccnt 0" ::: "memory");
#endif
  asm volatile("" ::: "memory");
}
#endif  // ASYNC_MODE

__device__ __forceinline__ float quad_dist(const float* pv, const float* tv) {
  float s = 0.0f;
#pragma unroll
  for (int j = 0; j < 4; ++j) {
    float dx = pv[3 * j + 0] - tv[3 * j + 0];
    float dy = pv[3 * j + 1] - tv[3 * j + 1];
    float dz = pv[3 * j + 2] - tv[3 * j + 2];
    s += sqrtf(fmaf(dz, dz, fmaf(dy, dy, dx * dx)));
  }
  return s;
}

__device__ __forceinline__ float block_reduce(float v, float* wsum, int tid) {
#pragma unroll
  for (int off = 16; off > 0; off >>= 1) v += __shfl_xor(v, off, 32);
  if ((tid & 31) == 0) wsum[tid >> 5] = v;
  __syncthreads();
  float t = 0.0f;
  if (tid == 0) {
#pragma unroll
    for (int w = 0; w < kNT / 32; ++w) t += wsum[w];
  }
  return t;
}

}  // namespace

__global__ void __launch_bounds__(kNT)
dis_loss_reduce_kernel(const float* __restrict__ pred,
                       const float* __restrict__ targ,
                       float* __restrict__ partial) {
  const int    tid       = threadIdx.x;
  const size_t blockBase = (size_t)blockIdx.x * (size_t)kCPB * kChunkFloats;
  float        acc       = 0.0f;

#if ASYNC_MODE
  // Double-buffered async global->LDS staging: 2 stages x (pred,target)
  // x 12 KB = 48 KB LDS (WGP has 320 KB -> ~6 blocks resident per WGP).
  __shared__ __align__(16) float sP[2][kChunkFloats];
  __shared__ __align__(16) float sT[2][kChunkFloats];
  const int lbase = tid * kFloatsPerThr;

  auto issue = [&](int c, int s) {
    const size_t off = blockBase + (size_t)c * kChunkFloats + (size_t)lbase;
    const float* gp = pred + off;
    const float* gt = targ + off;
    float*       lp = &sP[s][lbase];
    float*       lt = &sT[s][lbase];
    cp16(gp + 0, lp + 0); cp16(gp + 4, lp + 4); cp16(gp + 8, lp + 8);
    cp16(gt + 0, lt + 0); cp16(gt + 4, lt + 4); cp16(gt + 8, lt + 8);
  };

  issue(0, 0);
  for (int c = 0; c < kCPB; ++c) {
    const int s = c & 1;
    if (c + 1 < kCPB) {
      issue(c + 1, s ^ 1);  // prefetch next chunk into the other stage
      waitAsync6();         // in-order: the 6 oldest (chunk c) have landed
    } else {
      waitAsync0();
    }
    // Each lane consumes exactly the LDS region it staged -> no barriers.
    float pv[12], tv[12];
    const f32x4* p4 = (const f32x4*)&sP[s][lbase];
    const f32x4* t4 = (const f32x4*)&sT[s][lbase];
#pragma unroll
    for (int i = 0; i < 3; ++i) {
      *(f32x4*)&pv[4 * i] = p4[i];
      *(f32x4*)&tv[4 * i] = t4[i];
    }
    acc += quad_dist(pv, tv);
  }
#else
  // Fallback: direct 128-bit global loads (3 x b128 per input per chunk).
  for (int c = 0; c < kCPB; ++c) {
    const size_t off =
        blockBase + (size_t)c * kChunkFloats + (size_t)tid * kFloatsPerThr;
    const f32x4* p4 = (const f32x4*)(pred + off);
    const f32x4* t4 = (const f32x4*)(targ + off);
    float pv[12], tv[12];
#pragma unroll
    for (int i = 0; i < 3; ++i) {
      *(f32x4*)&pv[4 * i] = p4[i];
      *(f32x4*)&tv[4 * i] = t4[i];
    }
    acc += quad_dist(pv, tv);
  }
#endif

  __shared__ float wsum[kNT / 32];
  float total = block_reduce(acc, wsum, tid);
  if (tid == 0) partial[blockIdx.x] = total;
}

__global__ void __launch_bounds__(kNT)
dis_loss_final_kernel(const float* __restrict__ partial, float* __restrict__ out) {
  const int tid = threadIdx.x;
  float     s   = 0.0f;
  for (int i = tid; i < kNB; i += kNT) s += partial[i];
  __shared__ float wsum[kNT / 32];
  float total = block_reduce(s, wsum, tid);
  if (tid == 0) {
    constexpr float kInv = 1.0f / (float)kTriples;  // 5,505,024 < 2^24: exact
    out[0] = total * kInv;
  }
}

extern "C" void kernel_launch(void* const* d_in, const int* in_sizes, int n_in,
                              void* d_out, int out_size, void* d_ws,
                              size_t ws_size, hipStream_t stream) {
  (void)in_sizes; (void)n_in; (void)out_size; (void)ws_size;
  const float* pred    = (const float*)d_in[0];
  const float* targ    = (const float*)d_in[1];
  float*       partial = (float*)d_ws;   // kNB floats = 2688 B of scratch
  float*       out     = (float*)d_out;

  dis_loss_reduce_kernel<<<kNB, kNT, 0, stream>>>(pred, targ, partial);
  dis_loss_final_kernel<<<1, kNT, 0, stream>>>(partial, out);
}